// DecoderRNN_89893665505561
// MI455X (gfx1250) — compile-verified
//
#include <hip/hip_runtime.h>
#include <math.h>

typedef float v2f __attribute__((ext_vector_type(2)));
typedef float v8f __attribute__((ext_vector_type(8)));

#define V_  32000
#define E_  1024
#define H_  1024
#define S_  2048

__device__ __forceinline__ float sigm(float x) { return 1.0f / (1.0f + expf(-x)); }

__device__ __forceinline__ float wave_sum(float s) {
  #pragma unroll
  for (int off = 16; off > 0; off >>= 1) s += __shfl_down(s, off, 32);
  return s;
}

// ---------------------------------------------------------------- build x = [last_context, emb[word]]
__global__ void build_x_kernel(const float* __restrict__ last_ctx,
                               const float* __restrict__ emb,
                               const int*   __restrict__ word,
                               float* __restrict__ x) {
  int i = blockIdx.x * blockDim.x + threadIdx.x;           // 0..2047
  if (i >= H_ + E_) return;
  if (i < H_) x[i] = last_ctx[i];
  else        x[i] = emb[(size_t)word[0] * E_ + (i - H_)];
}

// ---------------------------------------------------------------- generic wave-per-row GEMV
// y[row] = act( dot(W[row*ld + coloff .. +K], x) + b[row] );  optional mirror write y2
__global__ __launch_bounds__(256)
void gemv_kernel(const float* __restrict__ W, int ld, int coloff, int K, int nrows,
                 const float* __restrict__ x, const float* __restrict__ b,
                 float* __restrict__ y, float* __restrict__ y2, int act) {
  int row  = blockIdx.x * (blockDim.x >> 5) + (threadIdx.x >> 5);
  if (row >= nrows) return;
  int lane = threadIdx.x & 31;
  const float4* w4 = (const float4*)(W + (size_t)row * ld + coloff);
  const float4* x4 = (const float4*)x;
  float s = 0.0f;
  int K4 = K >> 2;
  for (int i = lane; i < K4; i += 32) {
    float4 a = w4[i], c = x4[i];
    s = fmaf(a.x, c.x, s); s = fmaf(a.y, c.y, s);
    s = fmaf(a.z, c.z, s); s = fmaf(a.w, c.w, s);
  }
  s = wave_sum(s);
  if (lane == 0) {
    if (b)   s += b[row];
    if (act) s = tanhf(s);
    y[row] = s;
    if (y2) y2[row] = s;
  }
}

// ---------------------------------------------------------------- LSTM cell elementwise
__global__ void lstm_kernel(const float* __restrict__ ga, const float* __restrict__ gb,
                            const float* __restrict__ b_ih, const float* __restrict__ b_hh,
                            const float* __restrict__ c0,
                            float* __restrict__ ht_ws, float* __restrict__ ctxht,
                            float* __restrict__ out_ht, float* __restrict__ out_c) {
  int r = blockIdx.x * blockDim.x + threadIdx.x;           // 0..1023
  if (r >= H_) return;
  float gi = ga[r]        + gb[r]        + b_ih[r]        + b_hh[r];
  float gf = ga[H_ + r]   + gb[H_ + r]   + b_ih[H_ + r]   + b_hh[H_ + r];
  float gg = ga[2*H_ + r] + gb[2*H_ + r] + b_ih[2*H_ + r] + b_hh[2*H_ + r];
  float go = ga[3*H_ + r] + gb[3*H_ + r] + b_ih[3*H_ + r] + b_hh[3*H_ + r];
  float c  = sigm(gf) * c0[r] + sigm(gi) * tanhf(gg);
  float h  = sigm(go) * tanhf(c);
  ht_ws[r]      = h;
  ctxht[H_ + r] = h;   // second half of [context, ht] for W_ah GEMV
  out_ht[r]     = h;
  out_c[r]      = c;
}

// ---------------------------------------------------------------- attention scores via fp32 WMMA
// scores[s] = sum_j tanh( (enc @ Wa_e.T)[s,j] + u[j] ) * v[j]
// One block per 64 encoder rows (4 M-tiles); 8 waves, each owning 8 of the 64 N-tiles.
// Each K-step: 1 B-fragment load shared by 4 WMMAs (one per M-tile) -> 4x B reuse,
// Wa_e (4MB, L2-resident) is re-read only 32x instead of 128x.
__global__ __launch_bounds__(256)
void attn_scores_kernel(const float* __restrict__ enc,     // (2048,1024)
                        const float* __restrict__ W_attn,  // (1024,2048); Wa_e = cols 1024..2047
                        const float* __restrict__ u,       // (1024)
                        const float* __restrict__ v,       // (1024)
                        float* __restrict__ scores) {      // (2048)
  const int row0 = blockIdx.x * 64;                        // 32 blocks
  const int wave = threadIdx.x >> 5;
  const int lane = threadIdx.x & 31;
  const int l16  = lane & 15;
  const int hi   = lane >> 4;                              // 0: K+0..1, 1: K+2..3

  // A fragment sources: this lane's encoder rows for the 4 M-tiles
  const float* aRow = enc + (size_t)(row0 + l16) * E_ + 2 * hi;

  float part[4][8];
  #pragma unroll
  for (int m = 0; m < 4; ++m)
    #pragma unroll
    for (int r = 0; r < 8; ++r) part[m][r] = 0.0f;

  for (int jt = wave; jt < 64; jt += 8) {
    const int j0 = jt * 16;
    // B = Wa_e.T : B[k][j] = W_attn[j][H_ + k]; lane's N column = j0 + l16
    const float* bRow = W_attn + (size_t)(j0 + l16) * (E_ + H_) + H_ + 2 * hi;
    v8f acc[4];
    #pragma unroll
    for (int m = 0; m < 4; ++m) acc[m] = (v8f){0.f,0.f,0.f,0.f,0.f,0.f,0.f,0.f};

    for (int k = 0; k < E_; k += 4) {
      v2f b  = *(const v2f*)(bRow + k);
      v2f a0 = *(const v2f*)(aRow + k);
      v2f a1 = *(const v2f*)(aRow + 16 * E_ + k);
      v2f a2 = *(const v2f*)(aRow + 32 * E_ + k);
      v2f a3 = *(const v2f*)(aRow + 48 * E_ + k);
      acc[0] = __builtin_amdgcn_wmma_f32_16x16x4_f32(false, a0, false, b, (short)0, acc[0], false, false);
      acc[1] = __builtin_amdgcn_wmma_f32_16x16x4_f32(false, a1, false, b, (short)0, acc[1], false, false);
      acc[2] = __builtin_amdgcn_wmma_f32_16x16x4_f32(false, a2, false, b, (short)0, acc[2], false, false);
      acc[3] = __builtin_amdgcn_wmma_f32_16x16x4_f32(false, a3, false, b, (short)0, acc[3], false, false);
    }
    const int   j  = j0 + l16;
    const float uj = u[j], vj = v[j];
    #pragma unroll
    for (int m = 0; m < 4; ++m)
      #pragma unroll
      for (int r = 0; r < 8; ++r)                   // C: M = m*16 + r + 8*hi, N = j
        part[m][r] += tanhf(acc[m][r] + uj) * vj;
  }

  // reduce across the 16 lanes that share the same output rows
  #pragma unroll
  for (int m = 0; m < 4; ++m)
    #pragma unroll
    for (int r = 0; r < 8; ++r) {
      float s = part[m][r];
      #pragma unroll
      for (int off = 8; off > 0; off >>= 1) s += __shfl_xor(s, off, 16);
      part[m][r] = s;
    }
  __shared__ float red[8][4][16];
  if (l16 == 0) {
    #pragma unroll
    for (int m = 0; m < 4; ++m)
      #pragma unroll
      for (int r = 0; r < 8; ++r) red[wave][m][hi * 8 + r] = part[m][r];
  }
  __syncthreads();
  if (threadIdx.x < 64) {
    const int m  = threadIdx.x >> 4;
    const int rr = threadIdx.x & 15;
    float s = 0.0f;
    #pragma unroll
    for (int wv = 0; wv < 8; ++wv) s += red[wv][m][rr];
    scores[row0 + m * 16 + rr] = s;
  }
}

// ---------------------------------------------------------------- softmax over 2048 (one block)
__global__ __launch_bounds__(256)
void softmax_kernel(const float* __restrict__ scores,
                    float* __restrict__ w_ws, float* __restrict__ w_out) {
  __shared__ float sm[8];
  __shared__ float smax, ssum;
  int tid = threadIdx.x;
  float m = -3.402823466e38f;
  for (int i = tid; i < S_; i += 256) m = fmaxf(m, scores[i]);
  #pragma unroll
  for (int off = 16; off > 0; off >>= 1) m = fmaxf(m, __shfl_xor(m, off, 32));
  if ((tid & 31) == 0) sm[tid >> 5] = m;
  __syncthreads();
  if (tid == 0) { float t = sm[0]; for (int i = 1; i < 8; ++i) t = fmaxf(t, sm[i]); smax = t; }
  __syncthreads();
  float s = 0.0f;
  for (int i = tid; i < S_; i += 256) s += expf(scores[i] - smax);
  s = wave_sum(s);
  if ((tid & 31) == 0) sm[tid >> 5] = s;
  __syncthreads();
  if (tid == 0) { float t = 0.f; for (int i = 0; i < 8; ++i) t += sm[i]; ssum = t; }
  __syncthreads();
  for (int i = tid; i < S_; i += 256) {
    float ww = expf(scores[i] - smax) / ssum;
    w_ws[i] = ww; w_out[i] = ww;
  }
}

// ---------------------------------------------------------------- context[e] = sum_s w[s]*enc[s][e]
__global__ void context_kernel(const float* __restrict__ w, const float* __restrict__ enc,
                               float* __restrict__ ctxht) {
  int e = blockIdx.x * blockDim.x + threadIdx.x;           // 0..1023
  if (e >= E_) return;
  float s = 0.0f;
  for (int t = 0; t < S_; ++t) s = fmaf(w[t], enc[(size_t)t * E_ + e], s);
  ctxht[e] = s;
}

// ---------------------------------------------------------------- log-softmax reduce over 32000
__global__ __launch_bounds__(256)
void lsm_reduce_kernel(const float* __restrict__ logits, float* __restrict__ red) {
  __shared__ float sm[8];
  __shared__ float smax;
  int tid = threadIdx.x;
  float m = -3.402823466e38f;
  for (int i = tid; i < V_; i += 256) m = fmaxf(m, logits[i]);
  #pragma unroll
  for (int off = 16; off > 0; off >>= 1) m = fmaxf(m, __shfl_xor(m, off, 32));
  if ((tid & 31) == 0) sm[tid >> 5] = m;
  __syncthreads();
  if (tid == 0) { float t = sm[0]; for (int i = 1; i < 8; ++i) t = fmaxf(t, sm[i]); smax = t; }
  __syncthreads();
  float s = 0.0f;
  for (int i = tid; i < V_; i += 256) s += expf(logits[i] - smax);
  s = wave_sum(s);
  if ((tid & 31) == 0) sm[tid >> 5] = s;
  __syncthreads();
  if (tid == 0) {
    float t = 0.f;
    for (int i = 0; i < 8; ++i) t += sm[i];
    red[0] = smax; red[1] = logf(t);
  }
}

__global__ void lsm_write_kernel(const float* __restrict__ logits,
                                 const float* __restrict__ red,
                                 float* __restrict__ out) {
  int i = blockIdx.x * blockDim.x + threadIdx.x;
  if (i >= V_) return;
  out[i] = logits[i] - red[0] - red[1];
}

// ================================================================ launch
extern "C" void kernel_launch(void* const* d_in, const int* in_sizes, int n_in,
                              void* d_out, int out_size, void* d_ws, size_t ws_size,
                              hipStream_t stream) {
  const float* enc     = (const float*)d_in[0];
  const int*   word    = (const int*)  d_in[1];
  const float* lastctx = (const float*)d_in[2];
  const float* h0      = (const float*)d_in[3];
  const float* c0      = (const float*)d_in[4];
  const float* emb     = (const float*)d_in[5];
  const float* W_ih    = (const float*)d_in[6];
  const float* W_hh    = (const float*)d_in[7];
  const float* b_ih    = (const float*)d_in[8];
  const float* b_hh    = (const float*)d_in[9];
  const float* W_attn  = (const float*)d_in[10];
  const float* b_attn  = (const float*)d_in[11];
  const float* vvec    = (const float*)d_in[12];
  const float* W_ah    = (const float*)d_in[13];
  const float* b_ah    = (const float*)d_in[14];
  const float* W_out   = (const float*)d_in[15];
  const float* b_out   = (const float*)d_in[16];

  float* out = (float*)d_out;
  float* out_logp   = out;               // 32000
  float* out_ht     = out + 32000;       // 1024
  float* out_c      = out + 33024;       // 1024
  float* out_httil  = out + 34048;       // 1024
  float* out_w      = out + 35072;       // 2048

  float* ws = (float*)d_ws;              // all offsets 16B-aligned
  float* ws_x      = ws;                 // 2048
  float* ws_ga     = ws + 2048;          // 4096
  float* ws_gb     = ws + 6144;          // 4096
  float* ws_u      = ws + 10240;         // 1024
  float* ws_scores = ws + 11264;         // 2048
  float* ws_ht     = ws + 13312;         // 1024
  float* ws_w      = ws + 14336;         // 2048
  float* ws_ctxht  = ws + 16384;         // 2048  ([context, ht])
  float* ws_httil  = ws + 18432;         // 1024
  float* ws_logits = ws + 19456;         // 32000
  float* ws_red    = ws + 51456;         // 4

  // 1. x = [last_context, emb[word]]
  build_x_kernel<<<8, 256, 0, stream>>>(lastctx, emb, word, ws_x);
  // 2. gates = W_ih@x  and  W_hh@h0   (biases folded in the LSTM kernel)
  gemv_kernel<<<512, 256, 0, stream>>>(W_ih, 2048, 0, 2048, 4096, ws_x, nullptr, ws_ga, nullptr, 0);
  gemv_kernel<<<512, 256, 0, stream>>>(W_hh, 1024, 0, 1024, 4096, h0,   nullptr, ws_gb, nullptr, 0);
  // 3. LSTM cell -> ht, c
  lstm_kernel<<<4, 256, 0, stream>>>(ws_ga, ws_gb, b_ih, b_hh, c0, ws_ht, ws_ctxht, out_ht, out_c);
  // 4. u = Wa_h @ ht + b_attn   (cols 0..H-1 of W_attn)
  gemv_kernel<<<128, 256, 0, stream>>>(W_attn, 2048, 0, 1024, 1024, ws_ht, b_attn, ws_u, nullptr, 0);
  // 5. scores via fp32 WMMA GEMM fused with tanh/v-reduction
  attn_scores_kernel<<<32, 256, 0, stream>>>(enc, W_attn, ws_u, vvec, ws_scores);
  // 6. softmax -> w
  softmax_kernel<<<1, 256, 0, stream>>>(ws_scores, ws_w, out_w);
  // 7. context = w @ enc  -> first half of ws_ctxht
  context_kernel<<<4, 256, 0, stream>>>(ws_w, enc, ws_ctxht);
  // 8. ht_tilda = tanh(W_ah @ [context, ht] + b_ah)
  gemv_kernel<<<128, 256, 0, stream>>>(W_ah, 2048, 0, 2048, 1024, ws_ctxht, b_ah, ws_httil, out_httil, 1);
  // 9. logits = W_out @ ht_tilda + b_out
  gemv_kernel<<<4000, 256, 0, stream>>>(W_out, 1024, 0, 1024, 32000, ws_httil, b_out, ws_logits, nullptr, 0);
  // 10. log-softmax
  lsm_reduce_kernel<<<1, 256, 0, stream>>>(ws_logits, ws_red);
  lsm_write_kernel<<<125, 256, 0, stream>>>(ws_logits, ws_red, out_logp);
}